// SpUniformHeaderGraphAttentionLayer_61615600828791
// MI455X (gfx1250) — compile-verified
//
#include <hip/hip_runtime.h>
#include <math.h>

#define F_DIM 128
#define ALPHA 0.2f

typedef __attribute__((ext_vector_type(2))) float v2f;
typedef __attribute__((ext_vector_type(8))) float v8f;

// ---------------------------------------------------------------------------
// Kernel 0: zero the output accumulator (N*128 floats) and rowsum (N floats).
// ---------------------------------------------------------------------------
__global__ void gat_init_kernel(float* __restrict__ out, float* __restrict__ rowsum,
                                long outN, int N) {
    long i = (long)blockIdx.x * blockDim.x + threadIdx.x;
    if (i < outN) {
        out[i] = 0.0f;
    } else if (i < outN + N) {
        rowsum[i - outN] = 0.0f;
    }
}

// ---------------------------------------------------------------------------
// Kernel 1: projection  s_left = h @ a[0:128],  s_right = h @ a[128:256]
// via V_WMMA_F32_16X16X4_F32. One wave32 per 16-node tile, K swept 4 at a
// time (32 wmma issues per wave). B-matrix columns 0/1 hold a_left/a_right
// chunks, other 14 columns are zero; D columns 0/1 yield s_left/s_right.
//
// B is built BRANCHLESSLY: every lane issues the same (valid) loads from `a`
// (lane col>=2 reads col 0's address) and multiplies by a per-lane 0/1 scale.
// No exec-mask ladders -> loads clause up and overlap the WMMA chain.
//
// Layouts (ISA 7.12.2, wave32, 32-bit):
//   A (16x4):  lane L holds row M=L&15, K = (L>>4)*2 + {0,1} in v0,v1
//   B (4x16):  lane L holds col N=L&15, K = (L>>4)*2 + {0,1} in v0,v1
//   D (16x16): VGPR r, lanes 0-15 -> (M=r,   N=lane)
//              VGPR r, lanes 16-31-> (M=8+r, N=lane-16)
// ---------------------------------------------------------------------------
__global__ void gat_proj_wmma_kernel(const float* __restrict__ h,
                                     const float* __restrict__ a,
                                     float* __restrict__ sl,
                                     float* __restrict__ sr,
                                     int N) {
    const int lane = threadIdx.x & 31;
    const int wavesPerBlock = blockDim.x >> 5;
    const int tile = blockIdx.x * wavesPerBlock + (threadIdx.x >> 5);
    const int ntiles = (N + 15) >> 4;
    if (tile >= ntiles) return;           // wave-uniform exit, EXEC stays full

    const int base = tile << 4;
    const int m    = lane & 15;           // A row within tile / B,D column
    const int half = lane >> 4;
    const int koff = half * 2;            // this lane's K sub-offset {0 or 2}

    int row = base + m;
    if (row >= N) row = N - 1;            // clamp loads (stores guarded below)
    const float* hrow = h + (long)row * F_DIM + koff;

    // Branchless B setup: col1 reads a+128, everyone else reads a+0;
    // cols >= 2 get zeroed by `scale`.
    const float* aB = a + ((m == 1) ? F_DIM : 0) + koff;
    const float scale = (m < 2) ? 1.0f : 0.0f;

    v8f acc = {};
    #pragma unroll
    for (int k = 0; k < F_DIM; k += 4) {
        v2f A;
        A.x = hrow[k];
        A.y = hrow[k + 1];

        v2f B;
        B.x = scale * aB[k];
        B.y = scale * aB[k + 1];

        acc = __builtin_amdgcn_wmma_f32_16x16x4_f32(
            /*neg_a=*/false, A, /*neg_b=*/false, B,
            /*c_mod=*/(short)0, acc, /*reuse_a=*/false, /*reuse_b=*/false);
    }

    // Lane with column m==0 owns s_left for 8 rows, m==1 owns s_right.
    if (m < 2) {
        float* outp = (m == 0) ? sl : sr;
        #pragma unroll
        for (int r = 0; r < 8; ++r) {
            int rowOut = base + half * 8 + r;
            if (rowOut < N) outp[rowOut] = acc[r];
        }
    }
}

// ---------------------------------------------------------------------------
// Kernel 2: edge scatter. One wave32 per edge:
//   w = exp(-leaky_relu(sl[src] + sr[dst]))
//   rowsum[src] += w                      (lane 0)
//   out[src, :] += w * h[dst, :]          (each lane: float4 gather + 4 atomics)
// h (51.2MB) and the accumulator (51.2MB) both fit in the 192MB L2, so the
// random gathers and global_atomic_add_f32 RMWs stay on-chip.
// ---------------------------------------------------------------------------
__global__ void gat_edge_kernel(const float* __restrict__ h,
                                const int* __restrict__ src,
                                const int* __restrict__ dst,
                                const float* __restrict__ sl,
                                const float* __restrict__ sr,
                                float* __restrict__ out,
                                float* __restrict__ rowsum,
                                int E) {
    const int lane = threadIdx.x & 31;
    const long wave = (long)blockIdx.x * (blockDim.x >> 5) + (threadIdx.x >> 5);
    if (wave >= E) return;

    const int s = src[wave];
    const int d = dst[wave];

    float score = sl[s] + sr[d];                       // broadcast loads (L0 hit)
    float lr = score > 0.0f ? score : ALPHA * score;   // leaky_relu
    float w  = __expf(-lr);                            // bare v_exp_f32 path

    if (lane == 0) atomicAdd(rowsum + s, w);

    const float4 v = ((const float4*)(h + (long)d * F_DIM))[lane];  // 512B/wave gather
    float* op = out + (long)s * F_DIM + lane * 4;
    atomicAdd(op + 0, w * v.x);
    atomicAdd(op + 1, w * v.y);
    atomicAdd(op + 2, w * v.z);
    atomicAdd(op + 3, w * v.w);
}

// ---------------------------------------------------------------------------
// Kernel 3: out = relu(out / rowsum[node])
// ---------------------------------------------------------------------------
__global__ void gat_finalize_kernel(float* __restrict__ out,
                                    const float* __restrict__ rowsum,
                                    long outN) {
    long i = (long)blockIdx.x * blockDim.x + threadIdx.x;
    if (i >= outN) return;
    float v = out[i] / rowsum[i >> 7];   // F_DIM == 128
    out[i] = v > 0.0f ? v : 0.0f;
}

// ---------------------------------------------------------------------------
extern "C" void kernel_launch(void* const* d_in, const int* in_sizes, int n_in,
                              void* d_out, int out_size, void* d_ws, size_t ws_size,
                              hipStream_t stream) {
    const float* h    = (const float*)d_in[0];   // [N, 128]
    const float* a    = (const float*)d_in[1];   // [256]
    const int*   edge = (const int*)d_in[2];     // [2, E] (int32: JAX demotes int64)

    const int N = in_sizes[0] / F_DIM;
    const int E = in_sizes[2] / 2;
    const int* src = edge;
    const int* dst = edge + E;

    float* out    = (float*)d_out;               // [N, 128], used as accumulator
    float* sl     = (float*)d_ws;                // [N]
    float* sr     = sl + N;                      // [N]
    float* rowsum = sr + N;                      // [N]

    const int tpb = 256;                          // 8 waves per block (wave32)
    const long outN = (long)N * F_DIM;

    {   // zero accumulator + rowsum
        long tot = outN + N;
        int blocks = (int)((tot + tpb - 1) / tpb);
        gat_init_kernel<<<blocks, tpb, 0, stream>>>(out, rowsum, outN, N);
    }
    {   // WMMA projection
        int ntiles = (N + 15) / 16;
        int wavesPerBlock = tpb / 32;
        int blocks = (ntiles + wavesPerBlock - 1) / wavesPerBlock;
        gat_proj_wmma_kernel<<<blocks, tpb, 0, stream>>>(h, a, sl, sr, N);
    }
    {   // edge scatter: one wave per edge
        int wavesPerBlock = tpb / 32;
        int blocks = (E + wavesPerBlock - 1) / wavesPerBlock;
        gat_edge_kernel<<<blocks, tpb, 0, stream>>>(h, src, dst, sl, sr, out, rowsum, E);
    }
    {   // normalize + relu
        int blocks = (int)((outN + tpb - 1) / tpb);
        gat_finalize_kernel<<<blocks, tpb, 0, stream>>>(out, rowsum, outN);
    }
}